// AutomatonPELayer_82841329205907
// MI455X (gfx1250) — compile-verified
//
#include <hip/hip_runtime.h>
#include <hip/hip_bf16.h>

typedef __attribute__((ext_vector_type(2))) float v2f;
typedef __attribute__((ext_vector_type(8))) float v8f;

#define N_NODES   256
#define KDIM      16
#define ROWSTRIDE 20            // padded LDS row stride (floats): bank-conflict-free, 8B aligned
#define NROWS     (N_NODES + 2) // one zero pad row at each end (branch-free boundary)
#define ITERS     192           // rho(K) ~ 0.8997 -> residual ~1.6e-9, below fp32 eps
#define NTHREADS  512           // 16 wave32 waves, one per 16-node output tile

__global__ __launch_bounds__(NTHREADS, 1)
void automaton_pe_kernel(const float* __restrict__ p,   // [16]    pos_initial
                         const float* __restrict__ T,   // [16,16] pos_transition, row-major
                         float* __restrict__ out) {     // [256,16]
  __shared__ float X0[NROWS * ROWSTRIDE];
  __shared__ float X1[NROWS * ROWSTRIDE];

  const int tid  = threadIdx.x;
  const int lane = tid & 31;
  const int wave = tid >> 5;     // 0..15 -> node tile [16*wave, 16*wave+16)
  const int half = lane >> 4;    // lane group selects K/M sub-slice per ISA layout
  const int l15  = lane & 15;

  // ---- init: X0 interior rows = p (x^0 = v), pad rows = 0; X1 = 0 ----
  for (int idx = tid; idx < NROWS * ROWSTRIDE; idx += NTHREADS) {
    const int row = idx / ROWSTRIDE;
    const int col = idx % ROWSTRIDE;
    const bool interior = (row >= 1) && (row <= N_NODES) && (col < KDIM);
    X0[idx] = interior ? p[col] : 0.0f;
    X1[idx] = 0.0f;
  }

  // ---- constant WMMA operands ----
  // C tile: every output row accumulates p  =>  all 8 acc VGPRs = p[N=l15]
  const float pv = p[l15];
  // B = T^T split into 4 K-chunks of 4. ISA 32-bit 4x16 layout:
  // vgpr j, lane half h -> row K = 4*kk + 2*h + j, col N = l15;  B[K][N] = T[N][K]
  v2f Bop[4];
#pragma unroll
  for (int kk = 0; kk < 4; ++kk) {
    const int k0 = kk * 4 + 2 * half;
    Bop[kk].x = T[l15 * KDIM + k0];
    Bop[kk].y = T[l15 * KDIM + k0 + 1];
  }

  __syncthreads();

  // A operand addressing: lane reads neighbor rows of node (row0 + l15).
  // LDS row of node r is (r+1) because of the leading zero-pad row.
  const int row0 = wave * 16;
  const int rm1  = (row0 + l15) * ROWSTRIDE;      // node (row0+l15-1)
  const int rp1  = (row0 + l15 + 2) * ROWSTRIDE;  // node (row0+l15+1)

  for (int it = 0; it < ITERS; ++it) {
    const float* __restrict__ Xr = (it & 1) ? X1 : X0;
    float*       __restrict__ Xw = (it & 1) ? X0 : X1;

    v8f acc = {pv, pv, pv, pv, pv, pv, pv, pv};
#pragma unroll
    for (int kk = 0; kk < 4; ++kk) {
      // A 16x4 ISA layout: vgpr j, lane half h -> K = 4*kk + 2*h + j, M = l15
      const int k0 = kk * 4 + 2 * half;
      v2f a;
      a.x = Xr[rm1 + k0]     + Xr[rp1 + k0];
      a.y = Xr[rm1 + k0 + 1] + Xr[rp1 + k0 + 1];
      // D = A x B + C   (full fp32: v_wmma_f32_16x16x4_f32)
      acc = __builtin_amdgcn_wmma_f32_16x16x4_f32(
          false, a, false, Bop[kk], (short)0, acc, false, false);
    }

    // D 16x16 f32 layout: vgpr r, lane half h -> M = r + 8*h, N = l15
#pragma unroll
    for (int r = 0; r < 8; ++r) {
      const int node = row0 + r + 8 * half;
      Xw[(node + 1) * ROWSTRIDE + l15] = acc[r];
    }
    __syncthreads();  // one barrier per iteration (double buffering covers RAW+WAR)
  }

  const float* __restrict__ Xf = (ITERS & 1) ? X1 : X0;
  for (int idx = tid; idx < N_NODES * KDIM; idx += NTHREADS) {
    const int row = idx / KDIM;
    const int col = idx % KDIM;
    out[idx] = Xf[(row + 1) * ROWSTRIDE + col];
  }
}

extern "C" void kernel_launch(void* const* d_in, const int* in_sizes, int n_in,
                              void* d_out, int out_size, void* d_ws, size_t ws_size,
                              hipStream_t stream) {
  (void)in_sizes; (void)n_in; (void)d_ws; (void)ws_size; (void)out_size;
  const float* p = (const float*)d_in[0];   // pos_initial  [16]
  const float* T = (const float*)d_in[1];   // pos_transition [16,16]
  // d_in[2] = sentence_len (256, baked in as N_NODES)
  automaton_pe_kernel<<<1, NTHREADS, 0, stream>>>(p, T, (float*)d_out);
}